// TensorCPFactorization_44272522887177
// MI455X (gfx1250) — compile-verified
//
#include <hip/hip_runtime.h>
#include <stdint.h>

// ---------------------------------------------------------------------------
// CP-tensor query:  out[b] = sum_r f0[i0,r]*f1[i1,r]*f2[i2,r]*f3[i3,r]
//   DIM=512, RANK=256, ORDER=4, BATCH=100000, fp32.
// Memory-bound gather kernel: 400 MB of row gathers out of a 2 MB working set
// (lives in the 192 MB L2). Strategy: per-wave async global->LDS double
// buffering (CDNA5 ASYNCcnt path) + coalesced b128 row reads + wave32 tree
// reduction.
// ---------------------------------------------------------------------------

static constexpr int kRank          = 256;          // floats per factor row
static constexpr int kRowBytes      = kRank * 4;    // 1024 B
static constexpr int kOrder         = 4;
static constexpr int kBatchFloats   = kOrder * kRank;   // 1024 floats = 4 KB staged per query
static constexpr int kWavesPerBlock = 8;
static constexpr int kThreads       = kWavesPerBlock * 32;
static constexpr int kBlocks        = 2048;         // 16384 persistent waves

// Low 32 bits of a generic pointer to __shared__ == LDS byte offset
// (shared aperture occupies addr[63:32]).
__device__ __forceinline__ unsigned lds_off_u32(const void* p) {
  return (unsigned)(unsigned long long)p;
}

// Asynchronously copy one 1 KB factor row (256 fp32) into LDS.
// 32 lanes x 16 B x 2 issues = 1024 B. Tracked by ASYNCcnt (2 ops).
__device__ __forceinline__ void async_row_to_lds(const float* __restrict__ row,
                                                 unsigned ldsByteOff, int lane) {
  unsigned l0 = ldsByteOff + (unsigned)lane * 16u;
  unsigned long long g0 = (unsigned long long)row + (unsigned)lane * 16u;
  asm volatile(
      "global_load_async_to_lds_b128 %0, %2, off\n\t"
      "global_load_async_to_lds_b128 %1, %3, off"
      :
      : "v"(l0), "v"(l0 + 512u), "v"(g0), "v"(g0 + 512ull)
      : "memory");
}

// Stage all four gathered rows of query b into one 4 KB LDS buffer (8 async ops).
__device__ __forceinline__ void issue_query(const float* __restrict__ f0,
                                            const float* __restrict__ f1,
                                            const float* __restrict__ f2,
                                            const float* __restrict__ f3,
                                            const int*   __restrict__ idx,
                                            int b, unsigned bufOff, int lane) {
  const int4 iv = *(const int4*)(idx + 4ull * (unsigned)b);
  async_row_to_lds(f0 + (size_t)iv.x * kRank, bufOff + 0u * kRowBytes, lane);
  async_row_to_lds(f1 + (size_t)iv.y * kRank, bufOff + 1u * kRowBytes, lane);
  async_row_to_lds(f2 + (size_t)iv.z * kRank, bufOff + 2u * kRowBytes, lane);
  async_row_to_lds(f3 + (size_t)iv.w * kRank, bufOff + 3u * kRowBytes, lane);
}

__global__ __launch_bounds__(kThreads)
void cp_gather_dot_kernel(const float* __restrict__ f0,
                          const float* __restrict__ f1,
                          const float* __restrict__ f2,
                          const float* __restrict__ f3,
                          const int*   __restrict__ idx,
                          float* __restrict__ out,
                          int batch, int totalWaves) {
  // 8 waves * 2 ping-pong buffers * 4 KB = 64 KB LDS per workgroup.
  __shared__ float stage[kWavesPerBlock * 2 * kBatchFloats];

  const int lane  = (int)(threadIdx.x & 31u);
  const int wave  = (int)(threadIdx.x >> 5u);
  const int gwave = (int)blockIdx.x * kWavesPerBlock + wave;
  if (gwave >= batch) return;

  float* wbuf = &stage[(size_t)wave * 2 * kBatchFloats];
  const unsigned wbufOff = lds_off_u32(wbuf);

  int b   = gwave;
  int sel = 0;

  // Prologue: stage the first query.
  issue_query(f0, f1, f2, f3, idx, b, wbufOff, lane);

  while (b < batch) {
    const int  bn        = b + totalWaves;
    const bool have_next = (bn < batch);

    if (have_next) {
      // Pull the next index quad toward L2 and stage the next query's rows
      // into the other buffer while we compute on the current one.
      __builtin_prefetch(idx + 4ull * (unsigned)bn, 0, 1);
      issue_query(f0, f1, f2, f3, idx, bn, wbufOff + (unsigned)(sel ^ 1) * (kBatchFloats * 4u), lane);
      // 16 async ops outstanding; in-order completion => <=8 releases the
      // older buffer (current query).
      asm volatile("s_wait_asynccnt 0x8" ::: "memory");
    } else {
      asm volatile("s_wait_asynccnt 0x0" ::: "memory");
    }

    // ---- compute on current buffer: each lane owns 8 rank slots ----
    const float* buf = wbuf + sel * kBatchFloats;
    const float4* p0 = (const float4*)(buf + 0 * kRank);
    const float4* p1 = (const float4*)(buf + 1 * kRank);
    const float4* p2 = (const float4*)(buf + 2 * kRank);
    const float4* p3 = (const float4*)(buf + 3 * kRank);

    float acc = 0.0f;
#pragma unroll
    for (int c = 0; c < 2; ++c) {
      const int j = lane + c * 32;
      const float4 a = p0[j];
      const float4 x = p1[j];
      const float4 y = p2[j];
      const float4 z = p3[j];
      acc += a.x * x.x * y.x * z.x;
      acc += a.y * x.y * y.y * z.y;
      acc += a.z * x.z * y.z * z.z;
      acc += a.w * x.w * y.w * z.w;
    }

    // wave32 tree reduction (5 steps).
#pragma unroll
    for (int off = 16; off > 0; off >>= 1)
      acc += __shfl_xor(acc, off, 32);

    if (lane == 0) out[b] = acc;

    sel ^= 1;
    b = bn;
  }
}

extern "C" void kernel_launch(void* const* d_in, const int* in_sizes, int n_in,
                              void* d_out, int out_size, void* d_ws, size_t ws_size,
                              hipStream_t stream) {
  (void)in_sizes; (void)n_in; (void)d_ws; (void)ws_size;

  const float* f0  = (const float*)d_in[0];
  const float* f1  = (const float*)d_in[1];
  const float* f2  = (const float*)d_in[2];
  const float* f3  = (const float*)d_in[3];
  const int*   idx = (const int*)d_in[4];   // harness marshals ints as int32
  float* out = (float*)d_out;

  const int batch = out_size;               // 100000

  int blocks = kBlocks;
  const int maxBlocks = (batch + kWavesPerBlock - 1) / kWavesPerBlock;
  if (blocks > maxBlocks) blocks = maxBlocks;
  const int totalWaves = blocks * kWavesPerBlock;

  cp_gather_dot_kernel<<<blocks, kThreads, 0, stream>>>(
      f0, f1, f2, f3, idx, out, batch, totalWaves);
}